// SPResBlock_82471962018590
// MI455X (gfx1250) — compile-verified
//
#include <hip/hip_runtime.h>

// SPResBlock for gfx1250 (MI455X): fp32 WMMA path (V_WMMA_F32_16X16X4_F32).
// Pipeline: build grid -> neighbor table -> conv1(WMMA)+ReLU -> conv2(WMMA)+bias
//           +ReLU + residual GEMM(WMMA) + ReLU.
// - Branch-free gathers (mask multiply), neighbor indices preloaded per pair.
// - Double-buffered tap pipeline (fully unrolled): tap t+1's 32 gather loads
//   are issued before tap t's 128-WMMA chain (sched_barrier fenced).
// - Double-buffered LDS B-fragments: kc+1's 4 fragments load before kc's 8
//   WMMAs -> per-quad s_wait_dscnt stalls eliminated.
// - Two 16-row blocks per wave share every LDS B-fragment (halves ds reads).
// - 512-thread WGs; final output stored non-temporal (keep mid/features in L2).
// Workspace (~123 MB): [grid: 819200 i32][nbr: N*9 i32][mid: N*64 f32]

namespace {

constexpr int kB = 2, kH = 640, kW = 640;
constexpr int kN = 409600;
constexpr int kC = 64;
constexpr int kTaps = 9;
constexpr int kBlkRows = 16;
constexpr int kNBlk = kN / kBlkRows;   // 25600 row-blocks of 16 points
constexpr int kNPair = kNBlk / 2;      // 12800 block-pairs (2 blocks per wave)

typedef __attribute__((ext_vector_type(2))) float v2f;
typedef __attribute__((ext_vector_type(8))) float v8f;

__device__ __forceinline__ v8f wmma_f32_k4(v2f a, v2f b, v8f c) {
  // D(16x16,f32) = A(16x4,f32) x B(4x16,f32) + C
  return __builtin_amdgcn_wmma_f32_16x16x4_f32(false, a, false, b, (short)0, c,
                                               false, false);
}

__global__ void k_init_grid(int* __restrict__ grid, int n) {
  int i = blockIdx.x * blockDim.x + threadIdx.x;
  if (i < n) grid[i] = -1;
}

__global__ void k_scatter(const int* __restrict__ coords, int* __restrict__ grid) {
  int i = blockIdx.x * blockDim.x + threadIdx.x;
  if (i < kN) {
    int b = coords[3 * i + 0], y = coords[3 * i + 1], x = coords[3 * i + 2];
    grid[(b * kH + y) * kW + x] = i;
  }
}

__global__ void k_nbr(const int* __restrict__ coords, const int* __restrict__ grid,
                      int* __restrict__ nbr) {
  int i = blockIdx.x * blockDim.x + threadIdx.x;
  if (i >= kN) return;
  int b = coords[3 * i + 0], y = coords[3 * i + 1], x = coords[3 * i + 2];
#pragma unroll
  for (int t = 0; t < kTaps; ++t) {
    int ny = y + t / 3 - 1, nx = x + t % 3 - 1;
    int v = -1;
    if (ny >= 0 && ny < kH && nx >= 0 && nx < kW) v = grid[(b * kH + ny) * kW + nx];
    nbr[(long)i * kTaps + t] = v;
  }
}

// Issue the 16 A-fragment loads for one row-block of one tap (saddr + 32-bit
// voffset form: byte offsets fit in 32 bits for a 105MB feature tensor).
__device__ __forceinline__ void stage_tap(const char* __restrict__ featb, int idx,
                                          int rowfb, int h, float2 (&s)[16],
                                          float& m) {
  m = (idx >= 0) ? 1.f : 0.f;
  const int off = ((idx >= 0) ? idx : rowfb) << 8;  // row * 256 bytes
  const float2* ap = (const float2*)(featb + off) + h;
#pragma unroll
  for (int kc = 0; kc < 16; ++kc) s[kc] = ap[kc * 2];
}

// 128 WMMAs for one tap: 16 K-chunks x 4 N-tiles x 2 row-blocks, B shared and
// double-buffered in registers (next kc's fragments load before this kc's
// WMMAs -> no per-quad LDS-latency stalls).
__device__ __forceinline__ void compute_tap(const float2* __restrict__ wt,
                                            const float2 (&s0)[16], float m0,
                                            const float2 (&s1)[16], float m1,
                                            v8f (&a0)[4], v8f (&a1)[4]) {
  float2 bf[2][4];
#pragma unroll
  for (int nt = 0; nt < 4; ++nt) bf[0][nt] = wt[nt * 32];
#pragma unroll
  for (int kc = 0; kc < 16; ++kc) {
    if (kc < 15) {
#pragma unroll
      for (int nt = 0; nt < 4; ++nt)
        bf[(kc + 1) & 1][nt] = wt[(kc + 1) * 128 + nt * 32];
    }
    __builtin_amdgcn_sched_barrier(0);
    v2f A0;  A0.x = s0[kc].x * m0;  A0.y = s0[kc].y * m0;
    v2f A1;  A1.x = s1[kc].x * m1;  A1.y = s1[kc].y * m1;
#pragma unroll
    for (int nt = 0; nt < 4; ++nt) {
      v2f Bv;  Bv.x = bf[kc & 1][nt].x;  Bv.y = bf[kc & 1][nt].y;
      a0[nt] = wmma_f32_k4(A0, Bv, a0[nt]);
      a1[nt] = wmma_f32_k4(A1, Bv, a1[nt]);
    }
  }
}

// Dense 16x64 @ 64x64 tile GEMM (residual path), weights pre-shuffled in LDS.
__device__ __forceinline__ void dense_gemm_row(const char* __restrict__ resinb,
                                               const float* __restrict__ ldsd,
                                               int row, int h, int bofs,
                                               v8f (&rc)[4]) {
  const float2* ap = (const float2*)(resinb + ((long)row << 8)) + h;
  const float2* wt = (const float2*)ldsd + bofs;
  float2 st[16];
#pragma unroll
  for (int kc = 0; kc < 16; ++kc) st[kc] = ap[kc * 2];
  __builtin_amdgcn_sched_barrier(0);
  float2 bf[2][4];
#pragma unroll
  for (int nt = 0; nt < 4; ++nt) bf[0][nt] = wt[nt * 32];
#pragma unroll
  for (int kc = 0; kc < 16; ++kc) {
    if (kc < 15) {
#pragma unroll
      for (int nt = 0; nt < 4; ++nt)
        bf[(kc + 1) & 1][nt] = wt[(kc + 1) * 128 + nt * 32];
    }
    __builtin_amdgcn_sched_barrier(0);
    v2f A;  A.x = st[kc].x;  A.y = st[kc].y;
#pragma unroll
    for (int nt = 0; nt < 4; ++nt) {
      v2f Bv;  Bv.x = bf[kc & 1][nt].x;  Bv.y = bf[kc & 1][nt].y;
      rc[nt] = wmma_f32_k4(A, Bv, rc[nt]);
    }
  }
}

// One wave handles TWO 16-point blocks so each LDS B-fragment feeds 2 WMMAs.
// A-fragment per lane: row = lane&15, K-half = lane>>4 (16x4 f32 A layout).
template <bool RESIDUAL>
__global__ __launch_bounds__(512, 1) void k_conv(
    const float* __restrict__ feat,   // gathered input (N,64)
    const int* __restrict__ nbr,      // (N,9) neighbor indices, -1 = empty
    const float* __restrict__ w,      // (9,64,64) conv weights [tap][cin][cout]
    const float* __restrict__ bias,   // (64)
    const float* __restrict__ resin,  // residual input (N,64)      [RESIDUAL]
    const float* __restrict__ wd,     // (64,64) residual weights   [RESIDUAL]
    float* __restrict__ out) {
  extern __shared__ float lds[];
  const int tid = threadIdx.x;

  // Stage conv weights: lds[((t*16+kc)*64 + n)*4 + h*2 + j] = w[t][4kc+2h+j][n]
  for (int idx = tid; idx < kTaps * kC * kC; idx += blockDim.x) {
    int n = idx & 63, cin = (idx >> 6) & 63, t = idx >> 12;
    int kc = cin >> 2, hh = (cin >> 1) & 1, j = cin & 1;
    lds[((t * 16 + kc) * 64 + n) * 4 + hh * 2 + j] = w[idx];
  }
  if (RESIDUAL) {
    float* ldsd = lds + kTaps * kC * kC;
    for (int idx = tid; idx < kC * kC; idx += blockDim.x) {
      int n = idx & 63, cin = idx >> 6;
      int kc = cin >> 2, hh = (cin >> 1) & 1, j = cin & 1;
      ldsd[(kc * 64 + n) * 4 + hh * 2 + j] = wd[idx];
    }
  }
  __syncthreads();

  const int lane = tid & 31;
  const int wv = tid >> 5;
  const int nl = lane & 15;      // row (A) / col (B,C)
  const int h = lane >> 4;       // K-half (A,B) / M-half (C)
  const int bofs = nl * 2 + h;   // float2 offset inside one (t,kc) LDS row

  float bs[4];
#pragma unroll
  for (int nt = 0; nt < 4; ++nt) bs[nt] = bias[nt * 16 + nl];

  const char* fb = (const char*)feat;
  const float2* wbase = (const float2*)lds + bofs;

  const int wavesTotal = gridDim.x * (blockDim.x >> 5);
  for (int pr = blockIdx.x * (blockDim.x >> 5) + wv; pr < kNPair;
       pr += wavesTotal) {
    const int blk0 = pr * 2, blk1 = pr * 2 + 1;
    const int row0 = blk0 * kBlkRows + nl;
    const int row1 = blk1 * kBlkRows + nl;

    // Preload all 9+9 neighbor indices (contiguous per lane) up front.
    int nb0[kTaps], nb1[kTaps];
    {
      const int* np0 = nbr + (long)row0 * kTaps;
      const int* np1 = nbr + (long)row1 * kTaps;
#pragma unroll
      for (int t = 0; t < kTaps; ++t) { nb0[t] = np0[t]; nb1[t] = np1[t]; }
    }

    v8f ac0[4] = {{}, {}, {}, {}};
    v8f ac1[4] = {{}, {}, {}, {}};

    // Software-pipelined taps: stage tap t+1 while WMMA-ing tap t.
    float2 sA0[16], sA1[16], sB0[16], sB1[16];
    float mA0, mA1, mB0, mB1;
    stage_tap(fb, nb0[0], row0, h, sA0, mA0);
    stage_tap(fb, nb1[0], row1, h, sA1, mA1);
#pragma unroll
    for (int tp = 0; tp < 4; ++tp) {
      stage_tap(fb, nb0[2 * tp + 1], row0, h, sB0, mB0);
      stage_tap(fb, nb1[2 * tp + 1], row1, h, sB1, mB1);
      __builtin_amdgcn_sched_barrier(0);
      compute_tap(wbase + (2 * tp) * 2048, sA0, mA0, sA1, mA1, ac0, ac1);
      stage_tap(fb, nb0[2 * tp + 2], row0, h, sA0, mA0);
      stage_tap(fb, nb1[2 * tp + 2], row1, h, sA1, mA1);
      __builtin_amdgcn_sched_barrier(0);
      compute_tap(wbase + (2 * tp + 1) * 2048, sB0, mB0, sB1, mB1, ac0, ac1);
    }
    compute_tap(wbase + 8 * 2048, sA0, mA0, sA1, mA1, ac0, ac1);

    // Epilogue: bias + ReLU (+ residual GEMM + ReLU), then store.
    // C layout: VGPR r -> row (r + 8*h), col nl within each 16-wide tile.
#pragma unroll 1
    for (int half = 0; half < 2; ++half) {
      const int blk = half ? blk1 : blk0;
      const int row = half ? row1 : row0;
      v8f* ac = half ? ac1 : ac0;

      v8f rc[4] = {{}, {}, {}, {}};
      if (RESIDUAL)
        dense_gemm_row((const char*)resin, lds + kTaps * kC * kC, row, h, bofs,
                       rc);

      float* ob = out + (long)blk * kBlkRows * kC;
#pragma unroll
      for (int r = 0; r < 8; ++r) {
        float* op = ob + (r + 8 * h) * kC + nl;
#pragma unroll
        for (int nt = 0; nt < 4; ++nt) {
          float o = fmaxf(ac[nt][r] + bs[nt], 0.f);
          if (RESIDUAL) {
            o = fmaxf(o + rc[nt][r], 0.f);  // relu(relu(conv+b) + residual)
            __builtin_nontemporal_store(o, op + nt * 16);  // final out: NT
          } else {
            op[nt * 16] = o;  // mid: RT (re-read 9x by conv2, keep in L2)
          }
        }
      }
    }
  }
}

}  // namespace

extern "C" void kernel_launch(void* const* d_in, const int* in_sizes, int n_in,
                              void* d_out, int out_size, void* d_ws, size_t ws_size,
                              hipStream_t stream) {
  const float* features = (const float*)d_in[0];  // (N,64)
  const int* coords = (const int*)d_in[1];        // (N,3)
  const float* w1 = (const float*)d_in[2];        // (3,3,64,64)
  const float* b1 = (const float*)d_in[3];        // (64)
  const float* w2 = (const float*)d_in[4];        // (3,3,64,64)
  const float* b2 = (const float*)d_in[5];        // (64)
  const float* wd = (const float*)d_in[6];        // (64,64)
  float* out = (float*)d_out;

  int* grid = (int*)d_ws;                         //  3.28 MB
  int* nbr = grid + kB * kH * kW;                 // 14.75 MB
  float* mid = (float*)(nbr + (long)kN * kTaps);  // 104.86 MB

  k_init_grid<<<(kB * kH * kW + 255) / 256, 256, 0, stream>>>(grid, kB * kH * kW);
  k_scatter<<<(kN + 255) / 256, 256, 0, stream>>>(coords, grid);
  k_nbr<<<(kN + 255) / 256, 256, 0, stream>>>(coords, grid, nbr);

  const int convWGs = 400;  // 400 WGs x 16 waves -> 2 block-pairs per wave
  const size_t lds1 = (size_t)kTaps * kC * kC * sizeof(float);  // 147456 B
  const size_t lds2 = lds1 + (size_t)kC * kC * sizeof(float);   // 163840 B
  k_conv<false><<<convWGs, 512, lds1, stream>>>(features, nbr, w1, b1, nullptr,
                                                nullptr, mid);
  k_conv<true><<<convWGs, 512, lds2, stream>>>(mid, nbr, w2, b2, features, wd,
                                               out);
}